// BSpanDecoder_69209103007800
// MI455X (gfx1250) — compile-verified
//
#include <hip/hip_runtime.h>

// Problem constants (match reference)
constexpr int T = 128, B = 64, H = 512, E = 256, V = 32000;
constexpr int UNK = 2;
constexpr float EPS = 1e-10f;

typedef float v2f __attribute__((ext_vector_type(2)));
typedef float v8f __attribute__((ext_vector_type(8)));

__device__ __forceinline__ float sigmoidf(float x) { return 1.0f / (1.0f + __expf(-x)); }

#define WMMA_F32(Afrag, Bfrag, Cacc) \
    __builtin_amdgcn_wmma_f32_16x16x4_f32(false, (Afrag), false, (Bfrag), (short)0, (Cacc), false, false)

// ---------------------------------------------------------------------------
// WMMA-f32 GEMM specialized for M == 64 (all GEMMs here have 64 batch rows):
//   C[m][n] = sum_k A[m][k] * W[n][k]  (+ bias[n]),  m in [0,64)
// One wave per 16-column N-tile (grid = N/16). Each W float2 fragment is
// loaded ONCE and fed to 4 WMMAs (4 M-tiles, 4 accumulator chains) -> W
// stream traffic is the 1x minimum (proj_W = 131 MB is the HBM floor).
// Single-wave blocks: chip-wide parallelism comes from the 2000 blocks, so we
// relax waves/EU to let the allocator give every in-flight load its own VGPR
// pair (10 loads batched ahead of each 8-WMMA group -> staggered loadcnt
// waits instead of s_wait_loadcnt 0 serialization).
// V_WMMA_F32_16X16X4_F32 fragment layouts (ISA 7.12.2):
//   A 16x4:  lane = m (l15), half (lane>>4) selects K pair; vgpr j -> k = 2*half + j
//   B 4x16:  lane = n (l15), half selects K pair;           vgpr j -> k = 2*half + j
//   D 16x16: lane = n (l15), vgpr j -> m = 8*half + j
// ---------------------------------------------------------------------------
__global__ __launch_bounds__(32)
__attribute__((amdgpu_waves_per_eu(1, 2)))
void gemm_wmma_m64(
    const float* __restrict__ A, int lda,          // (64, K)
    const float* __restrict__ W, int ldw, int wcol,
    const float* __restrict__ bias,
    float* __restrict__ C, int ldc, int K)
{
    const int lane = threadIdx.x;
    const int l15  = lane & 15;
    const int half = lane >> 4;
    const int n0 = blockIdx.x * 16;

    const float* Wrow = W + (size_t)(n0 + l15) * ldw + wcol + 2 * half;
    const float* A0   = A + (size_t)l15 * lda + 2 * half;
    const size_t mstr = (size_t)16 * lda;

    v8f acc0 = {}, acc1 = {}, acc2 = {}, acc3 = {};
    for (int k0 = 0; k0 < K; k0 += 64) {
        // global_prefetch_b8: pull the streaming weight row ~2KB ahead
        __builtin_prefetch(Wrow + k0 + 512, 0, 1);
        #pragma unroll
        for (int kk = 0; kk < 64; kk += 8) {
            const int k = k0 + kk;
            // Batch all 10 loads (distinct registers) before the WMMA group.
            v2f w0  = *(const v2f*)(Wrow + k);
            v2f w1  = *(const v2f*)(Wrow + k + 4);
            v2f a00 = *(const v2f*)(A0 + k);
            v2f a01 = *(const v2f*)(A0 + mstr + k);
            v2f a02 = *(const v2f*)(A0 + 2 * mstr + k);
            v2f a03 = *(const v2f*)(A0 + 3 * mstr + k);
            v2f a10 = *(const v2f*)(A0 + k + 4);
            v2f a11 = *(const v2f*)(A0 + mstr + k + 4);
            v2f a12 = *(const v2f*)(A0 + 2 * mstr + k + 4);
            v2f a13 = *(const v2f*)(A0 + 3 * mstr + k + 4);
            acc0 = WMMA_F32(a00, w0, acc0);
            acc1 = WMMA_F32(a01, w0, acc1);
            acc2 = WMMA_F32(a02, w0, acc2);
            acc3 = WMMA_F32(a03, w0, acc3);
            acc0 = WMMA_F32(a10, w1, acc0);
            acc1 = WMMA_F32(a11, w1, acc1);
            acc2 = WMMA_F32(a12, w1, acc2);
            acc3 = WMMA_F32(a13, w1, acc3);
        }
    }
    const float bn = bias ? bias[n0 + l15] : 0.0f;
    float* Cp = C + (size_t)(8 * half) * ldc + n0 + l15;
    const size_t cstr = (size_t)16 * ldc;
    #pragma unroll
    for (int j = 0; j < 8; ++j) Cp[(size_t)j * ldc]            = acc0[j] + bn;
    #pragma unroll
    for (int j = 0; j < 8; ++j) Cp[cstr + (size_t)j * ldc]     = acc1[j] + bn;
    #pragma unroll
    for (int j = 0; j < 8; ++j) Cp[2 * cstr + (size_t)j * ldc] = acc2[j] + bn;
    #pragma unroll
    for (int j = 0; j < 8; ++j) Cp[3 * cstr + (size_t)j * ldc] = acc3[j] + bn;
}

// ---------------------------------------------------------------------------
// Fused encoder-score kernel (attention energy / copy score):
//   out[b][t] = sum_n dot(n) * tanh( sum_k enc[b,t,k]*W[n][wcol+k] + add(b,n) )
// enc is u_enc_out laid out (T,B,H); rows of a 16-tile share one b (T%16==0).
// One wave per 16-row tile; loops all N tiles of H, reduces dot over lanes.
// The 16x16 energy tile lives only in the accumulator VGPRs (never stored).
// ---------------------------------------------------------------------------
__global__ __launch_bounds__(32)
__attribute__((amdgpu_waves_per_eu(1, 2)))
void enc_score_wmma(
    const float* __restrict__ enc,        // (T,B,H)
    const float* __restrict__ W, int ldw, int wcol,
    const float* __restrict__ bias,       // per-n bias or nullptr
    const float* __restrict__ row_add,    // (B,H) per-(b,n) addend or nullptr
    const float* __restrict__ dotvec,     // (H) if !per_b else (B,H)
    int dot_per_b,
    float* __restrict__ out)              // (B,T)
{
    const int lane = threadIdx.x;
    const int l15  = lane & 15;
    const int half = lane >> 4;
    const int tile = blockIdx.x;          // 0 .. B*T/16-1
    const int b  = tile / (T / 16);
    const int t0 = (tile % (T / 16)) * 16;

    const float* Arow = enc + ((size_t)(t0 + l15) * B + b) * H + 2 * half;

    float racc[8] = {0.f, 0.f, 0.f, 0.f, 0.f, 0.f, 0.f, 0.f};

    for (int n0 = 0; n0 < H; n0 += 16) {
        const float* Wrow = W + (size_t)(n0 + l15) * ldw + wcol + 2 * half;
        v8f acc = {};
        for (int k0 = 0; k0 < H; k0 += 64) {
            __builtin_prefetch(Wrow + k0 + 512, 0, 1);
            #pragma unroll
            for (int kk = 0; kk < 64; kk += 8) {
                const int k = k0 + kk;
                v2f a0 = *(const v2f*)(Arow + k);
                v2f w0 = *(const v2f*)(Wrow + k);
                v2f a1 = *(const v2f*)(Arow + k + 4);
                v2f w1 = *(const v2f*)(Wrow + k + 4);
                acc = WMMA_F32(a0, w0, acc);
                acc = WMMA_F32(a1, w1, acc);
            }
        }
        const int n = n0 + l15;
        float add = 0.0f;
        if (row_add) add += row_add[(size_t)b * H + n];
        if (bias)    add += bias[n];
        const float dv = dot_per_b ? dotvec[(size_t)b * H + n] : dotvec[n];
        #pragma unroll
        for (int j = 0; j < 8; ++j)
            racc[j] += dv * tanhf(acc[j] + add);   // lane holds (m = 8*half+j, n)
    }
    // Sum over the 16 n-lanes within each half (xor masks stay inside a half).
    #pragma unroll
    for (int mask = 1; mask <= 8; mask <<= 1)
        #pragma unroll
        for (int j = 0; j < 8; ++j)
            racc[j] += __shfl_xor(racc[j], mask, 32);

    if (l15 == 0) {
        #pragma unroll
        for (int j = 0; j < 8; ++j)
            out[(size_t)b * T + t0 + 8 * half + j] = racc[j];
    }
}

// ---------------------------------------------------------------------------
// Embedding gather: xbuf[b][0:E] = emb[z[b]]
// ---------------------------------------------------------------------------
__global__ void embed_kernel(const int* __restrict__ z, const float* __restrict__ emb,
                             float* __restrict__ xbuf)
{
    int i = blockIdx.x * blockDim.x + threadIdx.x;   // B*E
    if (i < B * E) {
        int b = i / E, e = i % E;
        xbuf[(size_t)b * (E + H) + e] = emb[(size_t)z[b] * E + e];
    }
}

// ---------------------------------------------------------------------------
// Attention softmax over t + context:  alpha = softmax(rawA[b,:]),
// ctx[b,h] = sum_t alpha[t]*enc[b,t,h]; write ctx into x[:,E:] and y[:,H:].
// One block (128 threads) per b.
// ---------------------------------------------------------------------------
__global__ void attn_ctx_kernel(const float* __restrict__ enc,
                                const float* __restrict__ rawA,
                                float* __restrict__ xbuf, float* __restrict__ ybuf)
{
    const int b = blockIdx.x, tid = threadIdx.x;     // 128 threads == T
    __shared__ float sh[T];
    __shared__ float alpha[T];

    float v = rawA[(size_t)b * T + tid];
    sh[tid] = v; __syncthreads();
    for (int s = T / 2; s > 0; s >>= 1) {
        if (tid < s) sh[tid] = fmaxf(sh[tid], sh[tid + s]);
        __syncthreads();
    }
    const float mx = sh[0]; __syncthreads();
    float e = __expf(v - mx);
    sh[tid] = e; __syncthreads();
    for (int s = T / 2; s > 0; s >>= 1) {
        if (tid < s) sh[tid] += sh[tid + s];
        __syncthreads();
    }
    alpha[tid] = e / sh[0];
    __syncthreads();

    for (int h = tid; h < H; h += T) {
        float s = 0.0f;
        for (int t = 0; t < T; ++t)
            s += alpha[t] * enc[((size_t)t * B + b) * H + h];
        xbuf[(size_t)b * (E + H) + E + h] = s;
        ybuf[(size_t)b * (2 * H) + H + h] = s;
    }
}

// ---------------------------------------------------------------------------
// GRU gates:  h_new = (1-z)*tanh(i_n + r*h_n) + z*h, write h_new everywhere.
// ---------------------------------------------------------------------------
__global__ void gru_gate_kernel(const float* __restrict__ gi, const float* __restrict__ gh,
                                const float* __restrict__ h,
                                float* __restrict__ hnew, float* __restrict__ ybuf,
                                float* __restrict__ out_gru, float* __restrict__ out_hid)
{
    int i = blockIdx.x * blockDim.x + threadIdx.x;   // B*H
    if (i >= B * H) return;
    int b = i / H, n = i % H;
    const size_t gb = (size_t)b * 3 * H;
    float ir = gi[gb + n], iz = gi[gb + H + n], in = gi[gb + 2 * H + n];
    float hr = gh[gb + n], hz = gh[gb + H + n], hn = gh[gb + 2 * H + n];
    float r = sigmoidf(ir + hr);
    float z = sigmoidf(iz + hz);
    float nn = tanhf(in + r * hn);
    float hv = (1.0f - z) * nn + z * h[i];
    hnew[i] = hv;
    ybuf[(size_t)b * (2 * H) + n] = hv;
    out_gru[i] = hv;
    out_hid[i] = hv;
}

// ---------------------------------------------------------------------------
// Zero-fill (grid-stride)
// ---------------------------------------------------------------------------
__global__ void zero_kernel(float* __restrict__ p, size_t n)
{
    size_t i = (size_t)blockIdx.x * blockDim.x + threadIdx.x;
    size_t stride = (size_t)gridDim.x * blockDim.x;
    for (; i < n; i += stride) p[i] = 0.0f;
}

// ---------------------------------------------------------------------------
// Copy-mechanism stats + scatter:  ex=exp(rawC-cmax); total=sum(ex);
// scat[b][idx] += ex (mask: u_input!=0; idx==UNK -> V+t). One block per b.
// ---------------------------------------------------------------------------
__global__ void copy_scatter_kernel(const float* __restrict__ rawC,
                                    const int* __restrict__ u_input,   // (T,B)
                                    float* __restrict__ scat,          // (B, V+T)
                                    float* __restrict__ stats)         // (B, 2)
{
    const int b = blockIdx.x, t = threadIdx.x;   // T threads
    __shared__ float sh[T];
    float v = rawC[(size_t)b * T + t];
    sh[t] = v; __syncthreads();
    for (int s = T / 2; s > 0; s >>= 1) {
        if (t < s) sh[t] = fmaxf(sh[t], sh[t + s]);
        __syncthreads();
    }
    const float cmax = sh[0]; __syncthreads();
    float ex = __expf(v - cmax);
    sh[t] = ex; __syncthreads();
    for (int s = T / 2; s > 0; s >>= 1) {
        if (t < s) sh[t] += sh[t + s];
        __syncthreads();
    }
    if (t == 0) { stats[b * 2 + 0] = cmax; stats[b * 2 + 1] = sh[0]; }

    int u = u_input[(size_t)t * B + b];
    if (u != 0) {
        int idx = (u == UNK) ? (V + t) : u;
        atomicAdd(&scat[(size_t)b * (V + T) + idx], ex);
    }
}

// ---------------------------------------------------------------------------
// Final softmax over concat(gen_score[V], u_copy[V+T]) and proba assembly.
// u_copy[j] = log(EPS*total + (1-EPS)*scat[j]) + cmax   (computed on the fly).
// proba[j<V] = softmax(gen)[j] + softmax(ucopy)[j]; proba[V+t] = softmax(ucopy)[V+t].
// One block (256 threads) per b.
// ---------------------------------------------------------------------------
__global__ void final_softmax_kernel(const float* __restrict__ gen,   // (B,V)
                                     const float* __restrict__ scat,  // (B,V+T)
                                     const float* __restrict__ stats, // (B,2)
                                     float* __restrict__ proba)       // (B,V+T)
{
    const int b = blockIdx.x, tid = threadIdx.x;
    constexpr int NT = 256;
    constexpr int NTOT = 2 * V + T;
    const float cmax  = stats[b * 2 + 0];
    const float total = stats[b * 2 + 1];
    const float* genb  = gen  + (size_t)b * V;
    const float* scatb = scat + (size_t)b * (V + T);
    __shared__ float sh[NT];

    float mx = -3.4e38f;
    for (int i = tid; i < NTOT; i += NT) {
        float val = (i < V) ? genb[i]
                            : (logf(EPS * total + (1.0f - EPS) * scatb[i - V]) + cmax);
        mx = fmaxf(mx, val);
    }
    sh[tid] = mx; __syncthreads();
    for (int s = NT / 2; s > 0; s >>= 1) {
        if (tid < s) sh[tid] = fmaxf(sh[tid], sh[tid + s]);
        __syncthreads();
    }
    mx = sh[0]; __syncthreads();

    float sum = 0.0f;
    for (int i = tid; i < NTOT; i += NT) {
        float val = (i < V) ? genb[i]
                            : (logf(EPS * total + (1.0f - EPS) * scatb[i - V]) + cmax);
        sum += __expf(val - mx);
    }
    sh[tid] = sum; __syncthreads();
    for (int s = NT / 2; s > 0; s >>= 1) {
        if (tid < s) sh[tid] += sh[tid + s];
        __syncthreads();
    }
    const float inv = 1.0f / sh[0];

    float* pb = proba + (size_t)b * (V + T);
    for (int j = tid; j < V + T; j += NT) {
        float uc = logf(EPS * total + (1.0f - EPS) * scatb[j]) + cmax;
        float p = __expf(uc - mx) * inv;
        if (j < V) p += __expf(genb[j] - mx) * inv;
        pb[j] = p;
    }
}

// ===========================================================================
extern "C" void kernel_launch(void* const* d_in, const int* in_sizes, int n_in,
                              void* d_out, int out_size, void* d_ws, size_t ws_size,
                              hipStream_t stream)
{
    (void)in_sizes; (void)n_in; (void)out_size; (void)ws_size;

    const float* u_enc_out = (const float*)d_in[0];   // (T,B,H)
    const int*   z_tm1     = (const int*)  d_in[1];   // (1,B)
    const float* last_hid  = (const float*)d_in[2];   // (1,B,H)
    const int*   u_input   = (const int*)  d_in[3];   // (T,B)
    const float* emb       = (const float*)d_in[4];   // (V,E)
    const float* attn_W    = (const float*)d_in[5];   // (H,2H)
    const float* attn_b    = (const float*)d_in[6];   // (H)
    const float* attn_v    = (const float*)d_in[7];   // (H)
    const float* gru_Wih   = (const float*)d_in[8];   // (3H, H+E)
    const float* gru_Whh   = (const float*)d_in[9];   // (3H, H)
    const float* gru_bih   = (const float*)d_in[10];  // (3H)
    const float* gru_bhh   = (const float*)d_in[11];  // (3H)
    const float* proj_W    = (const float*)d_in[12];  // (V, 2H)
    const float* proj_b    = (const float*)d_in[13];  // (V)
    const float* copy1_W   = (const float*)d_in[14];  // (H,H)
    const float* copy1_b   = (const float*)d_in[15];  // (H)

    float* out = (float*)d_out;
    float* out_gru   = out;                     // (B,H)
    float* out_hid   = out + (size_t)B * H;     // (B,H)
    float* out_proba = out + (size_t)2 * B * H; // (B, V+T)

    // Workspace layout (floats), ~18 MB total
    float* ws    = (float*)d_ws;
    float* xbuf  = ws;                          // (B, E+H) = concat(emb, ctx)
    float* ybuf  = xbuf + (size_t)B * (E + H);  // (B, 2H)  = concat(h_new, ctx)
    float* hW    = ybuf + (size_t)B * 2 * H;    // (B, H)   h@W_h^T + attn_b
    float* rawA  = hW   + (size_t)B * H;        // (B, T)
    float* gi    = rawA + (size_t)B * T;        // (B, 3H)
    float* gh    = gi   + (size_t)B * 3 * H;    // (B, 3H)
    float* hnew  = gh   + (size_t)B * 3 * H;    // (B, H)
    float* rawC  = hnew + (size_t)B * H;        // (B, T)
    float* stats = rawC + (size_t)B * T;        // (B, 2)
    float* gen   = stats + (size_t)B * 2;       // (B, V)
    float* scat  = gen  + (size_t)B * V;        // (B, V+T)

    const float* h = last_hid;                  // (B,H)

    // 1) x[:, :E] = emb[z]
    embed_kernel<<<(B * E + 255) / 256, 256, 0, stream>>>(z_tm1, emb, xbuf);

    // 2) hW = h @ attn_W[:, :H]^T + attn_b   (fold attn_b into the row addend)
    gemm_wmma_m64<<<H / 16, 32, 0, stream>>>(
        h, H, attn_W, 2 * H, 0, attn_b, hW, H, H);

    // 3) rawA[b,t] = sum_n attn_v[n]*tanh(enc@W_e^T + hW)   (W_e = attn_W[:, H:])
    enc_score_wmma<<<(B * T) / 16, 32, 0, stream>>>(
        u_enc_out, attn_W, 2 * H, H, nullptr, hW, attn_v, 0, rawA);

    // 4) alpha = softmax_t(rawA); ctx -> x[:,E:], y[:,H:]
    attn_ctx_kernel<<<B, T, 0, stream>>>(u_enc_out, rawA, xbuf, ybuf);

    // 5) gi = x @ Wih^T + bih   (K = E+H = 768)
    gemm_wmma_m64<<<3 * H / 16, 32, 0, stream>>>(
        xbuf, E + H, gru_Wih, E + H, 0, gru_bih, gi, 3 * H, E + H);

    // 6) gh = h @ Whh^T + bhh
    gemm_wmma_m64<<<3 * H / 16, 32, 0, stream>>>(
        h, H, gru_Whh, H, 0, gru_bhh, gh, 3 * H, H);

    // 7) GRU gates -> h_new (also y[:, :H], out_gru, out_hid)
    gru_gate_kernel<<<(B * H + 255) / 256, 256, 0, stream>>>(
        gi, gh, h, hnew, ybuf, out_gru, out_hid);

    // 8) gen = y @ proj_W^T + proj_b  (M=64, N=32000, K=1024; proj_W streamed 1x)
    gemm_wmma_m64<<<V / 16, 32, 0, stream>>>(
        ybuf, 2 * H, proj_W, 2 * H, 0, proj_b, gen, V, 2 * H);

    // 9) rawC[b,t] = sum_n h_new[b,n]*tanh(enc@copy1_W^T + copy1_b)
    enc_score_wmma<<<(B * T) / 16, 32, 0, stream>>>(
        u_enc_out, copy1_W, H, 0, copy1_b, nullptr, hnew, 1, rawC);

    // 10) scat = 0
    zero_kernel<<<512, 256, 0, stream>>>(scat, (size_t)B * (V + T));

    // 11) copy stats + scatter
    copy_scatter_kernel<<<B, T, 0, stream>>>(rawC, u_input, scat, stats);

    // 12) final softmax over [gen | u_copy] and proba assembly
    final_softmax_kernel<<<B, 256, 0, stream>>>(gen, scat, stats, out_proba);
}